// GRUPolicy_60559038873742
// MI455X (gfx1250) — compile-verified
//
#include <hip/hip_runtime.h>

#define B_   2048
#define T_   512
#define V_   4
#define E_   64
#define H_   128
#define G_   384      // 3*H
#define BM   16       // batch rows per workgroup
#define HSTR 132      // padded LDS row stride (floats) -> conflict-free b64 A loads

typedef float v2f __attribute__((ext_vector_type(2)));
typedef float v8f __attribute__((ext_vector_type(8)));

__device__ __forceinline__ float fsig(float x)  { return 1.0f / (1.0f + __expf(-x)); }
__device__ __forceinline__ float ftanh(float x) { return 2.0f / (1.0f + __expf(-2.0f * x)) - 1.0f; }

// ---------------------------------------------------------------------------
// Stage 1: Gtab[v][g] = b_ih[g] + dot(emb[v,:], W_ih[g,:])   (only 4 tokens!)
// ---------------------------------------------------------------------------
__global__ void gru_gitab_kernel(const float* __restrict__ emb,
                                 const float* __restrict__ W_ih,
                                 const float* __restrict__ b_ih,
                                 float* __restrict__ Gtab) {
  int g = threadIdx.x;
  if (g >= G_) return;
  float s0 = b_ih[g], s1 = b_ih[g], s2 = b_ih[g], s3 = b_ih[g];
  #pragma unroll 8
  for (int e = 0; e < E_; ++e) {
    float w = W_ih[g * E_ + e];
    s0 += emb[0 * E_ + e] * w;
    s1 += emb[1 * E_ + e] * w;
    s2 += emb[2 * E_ + e] * w;
    s3 += emb[3 * E_ + e] * w;
  }
  Gtab[0 * G_ + g] = s0;
  Gtab[1 * G_ + g] = s1;
  Gtab[2 * G_ + g] = s2;
  Gtab[3 * G_ + g] = s3;
}

// ---------------------------------------------------------------------------
// Stage 2: time-scan. 1 block = 16 batch rows, 8 waves.
// Wave w owns hidden columns [16w, 16w+16): computes r/z/n tiles via WMMA f32,
// weights held in VGPRs, h double-buffered in LDS, h_prev loop-carried in regs.
// ---------------------------------------------------------------------------
__global__ __launch_bounds__(256)
void gru_scan_kernel(const int* __restrict__ x,
                     const float* __restrict__ W_hh,
                     const float* __restrict__ b_hh,
                     const float* __restrict__ W_fc,
                     const float* __restrict__ b_fc,
                     const float* __restrict__ Gtab,
                     float* __restrict__ out) {
  __shared__ float hbuf[2][BM * HSTR];        // double-buffered hidden state
  __shared__ float gtab[V_ * G_];             // token -> gi table
  __shared__ unsigned char toks[BM][T_];      // token ids for this row tile

  const int tid  = threadIdx.x;
  const int lane = tid & 31;
  const int wv   = tid >> 5;        // wave 0..7
  const int c    = lane & 15;       // column / A-row within tile
  const int hi   = lane >> 4;       // K-half / M-half selector
  const int rb   = blockIdx.x * BM; // batch row base

  // ---- preload tokens (values 0..3 -> bytes) ----
  for (int idx = tid; idx < BM * T_; idx += 256) {
    int m = idx >> 9;          // / T_
    int t = idx & (T_ - 1);
    toks[m][t] = (unsigned char)x[(rb + m) * T_ + t];
  }
  // ---- preload gi table ----
  for (int idx = tid; idx < V_ * G_; idx += 256) gtab[idx] = Gtab[idx];
  // ---- zero hidden state buffers ----
  for (int idx = tid; idx < BM * HSTR; idx += 256) {
    hbuf[0][idx] = 0.0f;
    hbuf[1][idx] = 0.0f;
  }

  // ---- per-lane constants ----
  const int hcol = wv * 16 + c;               // hidden column 0..127
  const float bR = b_hh[hcol];
  const float bZ = b_hh[H_ + hcol];
  const float bN = b_hh[2 * H_ + hcol];

  // ---- preload W_hh B-operands into registers: 3 gate tiles x 32 k-steps ----
  // B[k][n] = W_hh[n][k]; lane layout: VGPR0 = rows {2*hi}, VGPR1 = {2*hi+1}
  v2f bw[3][32];
  #pragma unroll
  for (int g = 0; g < 3; ++g) {
    const float* wp = W_hh + (g * H_ + hcol) * H_ + 2 * hi;
    #pragma unroll
    for (int k = 0; k < 32; ++k)
      bw[g][k] = *(const v2f*)(wp + 4 * k);
  }

  float hreg[8];                               // this lane's h_prev slice
  #pragma unroll
  for (int i = 0; i < 8; ++i) hreg[i] = 0.0f;

  __syncthreads();

  int rbuf = 0;
  for (int t = 0; t < T_; ++t) {
    const float* hr = hbuf[rbuf];
    v8f accR = {}; v8f accZ = {}; v8f accN = {};
    const float* ap = hr + c * HSTR + 2 * hi;  // A: h[row=c][k-half=hi]
    #pragma unroll
    for (int k = 0; k < 32; ++k) {
      v2f a = *(const v2f*)(ap + 4 * k);
      accR = __builtin_amdgcn_wmma_f32_16x16x4_f32(false, a, false, bw[0][k],
                                                   (short)0, accR, false, false);
      accZ = __builtin_amdgcn_wmma_f32_16x16x4_f32(false, a, false, bw[1][k],
                                                   (short)0, accZ, false, false);
      accN = __builtin_amdgcn_wmma_f32_16x16x4_f32(false, a, false, bw[2][k],
                                                   (short)0, accN, false, false);
    }
    // ---- gate math + h update (D layout: VGPR i -> row i + 8*hi, col c) ----
    float* hw = hbuf[rbuf ^ 1];
    #pragma unroll
    for (int i = 0; i < 8; ++i) {
      int m = i + 8 * hi;
      int tok = toks[m][t];
      const float* gt = gtab + tok * G_ + hcol;
      float r  = fsig (gt[0]      + accR[i] + bR);
      float z  = fsig (gt[H_]     + accZ[i] + bZ);
      float n  = ftanh(gt[2 * H_] + r * (accN[i] + bN));
      float hn = n + z * (hreg[i] - n);
      hreg[i] = hn;
      hw[m * HSTR + hcol] = hn;
    }
    __syncthreads();
    rbuf ^= 1;
  }

  // ---- final FC: logits[b][v] = hT . W_fc[v] + b_fc[v]  (V=4) ----
  // after 512 toggles rbuf==0, which holds the last-written state.
  if (tid < BM * V_) {
    int row = tid >> 2;
    int v   = tid & 3;
    float acc = b_fc[v];
    const float* hf = hbuf[rbuf] + row * HSTR;
    const float* wf = W_fc + v * H_;
    #pragma unroll 8
    for (int k = 0; k < H_; ++k) acc += hf[k] * wf[k];
    out[(rb + row) * V_ + v] = acc;
  }
}

// ---------------------------------------------------------------------------
extern "C" void kernel_launch(void* const* d_in, const int* in_sizes, int n_in,
                              void* d_out, int out_size, void* d_ws, size_t ws_size,
                              hipStream_t stream) {
  (void)in_sizes; (void)n_in; (void)out_size; (void)ws_size;
  const int*   x    = (const int*)d_in[0];
  const float* emb  = (const float*)d_in[1];
  const float* W_ih = (const float*)d_in[2];
  const float* W_hh = (const float*)d_in[3];
  const float* b_ih = (const float*)d_in[4];
  const float* b_hh = (const float*)d_in[5];
  const float* W_fc = (const float*)d_in[6];
  const float* b_fc = (const float*)d_in[7];
  float* out  = (float*)d_out;
  float* Gtab = (float*)d_ws;   // 4*384 floats

  gru_gitab_kernel<<<1, G_, 0, stream>>>(emb, W_ih, b_ih, Gtab);
  gru_scan_kernel<<<B_ / BM, 256, 0, stream>>>(x, W_hh, b_hh, W_fc, b_fc, Gtab, out);
}